// Vanilla_Backproj_49709951484538
// MI455X (gfx1250) — compile-verified
//
#include <hip/hip_runtime.h>
#include <cstdint>
#include <math.h>

// ---- problem constants (from reference) ------------------------------------
#define N_ANGLES 1000
#define N_DET    513
#define IMG      512
#define CROP     362
#define TOP      75                 // (512 - 362) / 2
#define CHUNK    10                 // angles staged per TDM transfer
#define NCHUNK   (N_ANGLES / CHUNK) // 100
#define TILE     32                 // 32x32 pixel tile, 256 thr, 4 px/thread

typedef __attribute__((ext_vector_type(4))) unsigned int v4u;
typedef __attribute__((ext_vector_type(8))) unsigned int v8u;

// ---- Tensor Data Mover: load CHUNK*N_DET contiguous floats into LDS --------
// D# per CDNA5 ISA. 2-SGPR-group form (2D tensor,
// VADDR2/VADDR3 NULL). Tracked by TENSORcnt.
__device__ __forceinline__ void tdm_load_chunk(const float* gbase,
                                               unsigned lds_byte_addr,
                                               unsigned long long elem_off) {
    unsigned long long gaddr =
        (unsigned long long)(uintptr_t)gbase + elem_off * 4ull;

    const unsigned td0   = (unsigned)(N_ANGLES * N_DET); // 513000 elems/batch
    const unsigned tile0 = (unsigned)(CHUNK * N_DET);    // 5130 contiguous

    v4u g0;
    g0.x = 1u;                                  // count=1 (valid user D#)
    g0.y = lds_byte_addr;                       // LDS byte address of tile
    g0.z = (unsigned)(gaddr & 0xFFFFFFFFu);     // global_addr[31:0]
    g0.w = (unsigned)(gaddr >> 32)              // global_addr[56:32]
         | 0x80000000u;                         // type=2 ("image") [127:126]

    v8u g1;
    g1.s0 = 2u << 16;                           // data_size=2 -> 4 bytes
    g1.s1 = (td0 & 0xFFFFu) << 16;              // tensor_dim0[15:0]
    g1.s2 = (td0 >> 16) | (1u << 16);           // tensor_dim0[31:16] | tensor_dim1=1
    g1.s3 = tile0 << 16;                        // tile_dim0
    g1.s4 = 1u;                                 // tile_dim1=1, tile_dim2=0
    g1.s5 = td0;                                // tensor_dim0_stride[31:0]
    g1.s6 = 0u;                                 // stride hi / dim1_stride lo
    g1.s7 = 0u;

    asm volatile("tensor_load_to_lds %0, %1" :: "s"(g0), "s"(g1) : "memory");
}

// ---- one interpolated sample accumulated -----------------------------------
__device__ __forceinline__ float interp_acc(const float* __restrict__ row,
                                            float p, float acc) {
    float fp = floorf(p);
    float w  = p - fp;
    int   i0 = (int)fp;
    // nested ternary -> single v_med3_i32; never triggers for in-crop pixels,
    // only keeps the edge-tile ghost pixels' LDS reads in range
    i0 = i0 < 0 ? 0 : (i0 > (N_DET - 2) ? (N_DET - 2) : i0);
    float v0 = row[i0];
    float v1 = row[i0 + 1];
    return fmaf(w, v1 - v0, acc + v0);          // acc + v0*(1-w) + v1*w
}

__global__ __launch_bounds__(256)
void backproj_tdm_kernel(const float* __restrict__ x, float* __restrict__ out) {
    __shared__ float2 s_trig[N_ANGLES];         // interleaved {cos, sin}: one
                                                // ds_load_b64 per angle
    __shared__ float  s_buf[2][CHUNK * N_DET];  // double-buffered angle chunks

    const int b   = blockIdx.z;
    const int oib = blockIdx.y * TILE;          // output dim-2 (x-coordinate)
    const int ojb = blockIdx.x * TILE;          // output dim-3 (y-coordinate)
    const int li  = (int)(threadIdx.x >> 4);    // 0..15
    const int lj  = (int)(threadIdx.x & 15);    // 0..15 (lane-fast -> oj)

    // wave-uniform scalar wave id -> guarantees scalar branch around TDM issue
    const int wid = __builtin_amdgcn_readfirstlane((int)threadIdx.x) >> 5;

    const unsigned lds_b0 =
        (unsigned)(unsigned long long)(uintptr_t)&s_buf[0][0];
    const unsigned lds_b1 =
        (unsigned)(unsigned long long)(uintptr_t)&s_buf[1][0];
    const unsigned long long batch_off = (unsigned long long)b *
                                         (unsigned long long)(N_ANGLES * N_DET);

    // prologue: kick chunk 0 DMA, build trig table while it flies
    if (wid == 0) tdm_load_chunk(x, lds_b0, batch_off);

    const float kth = 3.14159265358979323846f / (float)N_ANGLES;
    for (int a = (int)threadIdx.x; a < N_ANGLES; a += 256) {
        float sv, cv;
        __sincosf((float)a * kth, &sv, &cv);
        s_trig[a] = make_float2(cv, sv);
    }

    // pixel coordinates scaled so that p = xs*cos + ys*sin + 256
    const float step = 0.26f / (float)(IMG - 1);       // linspace step
    const float invD = (float)(IMG) / 0.26f;           // 1/delta_s
    const float xs0 = (-0.13f + (float)(TOP + oib + li)      * step) * invD;
    const float xs1 = (-0.13f + (float)(TOP + oib + li + 16) * step) * invD;
    const float ys0 = (-0.13f + (float)(TOP + ojb + lj)      * step) * invD;
    const float ys1 = (-0.13f + (float)(TOP + ojb + lj + 16) * step) * invD;

    if (wid == 0) __builtin_amdgcn_s_wait_tensorcnt(0);
    __syncthreads();                                    // chunk 0 visible

    float acc00 = 0.f, acc01 = 0.f, acc10 = 0.f, acc11 = 0.f;

    for (int k = 0; k < NCHUNK; ++k) {
        // issue next chunk into the free buffer (freed by last barrier)
        if (wid == 0 && (k + 1) < NCHUNK) {
            tdm_load_chunk(x, (k & 1) ? lds_b0 : lds_b1,
                           batch_off +
                           (unsigned long long)(k + 1) * (CHUNK * N_DET));
        }

        const float* __restrict__ buf = s_buf[k & 1];
        const int abase = k * CHUNK;
#pragma unroll
        for (int j = 0; j < CHUNK; ++j) {
            const float2 cs = s_trig[abase + j];        // one ds_load_b64
            const float c = cs.x, s = cs.y;
            const float* __restrict__ row = buf + j * N_DET;
            const float u0 = fmaf(xs0, c, 256.0f);
            const float u1 = fmaf(xs1, c, 256.0f);
            acc00 = interp_acc(row, fmaf(ys0, s, u0), acc00);
            acc01 = interp_acc(row, fmaf(ys1, s, u0), acc01);
            acc10 = interp_acc(row, fmaf(ys0, s, u1), acc10);
            acc11 = interp_acc(row, fmaf(ys1, s, u1), acc11);
        }

        if (wid == 0) __builtin_amdgcn_s_wait_tensorcnt(0); // next chunk landed
        __syncthreads();                                    // publish + free buf
    }

    // out[b][0][i][j] = acc(x=lin[TOP+i], y=lin[TOP+j]) * pi/N_ANGLES
    const float scale = 3.14159265358979323846f / (float)N_ANGLES;
    const int oi0 = oib + li, oi1 = oib + li + 16;
    const int oj0 = ojb + lj, oj1 = ojb + lj + 16;
    float* ob = out + (size_t)b * CROP * CROP;
    if (oi0 < CROP && oj0 < CROP) ob[oi0 * CROP + oj0] = acc00 * scale;
    if (oi0 < CROP && oj1 < CROP) ob[oi0 * CROP + oj1] = acc01 * scale;
    if (oi1 < CROP && oj0 < CROP) ob[oi1 * CROP + oj0] = acc10 * scale;
    if (oi1 < CROP && oj1 < CROP) ob[oi1 * CROP + oj1] = acc11 * scale;
}

extern "C" void kernel_launch(void* const* d_in, const int* in_sizes, int n_in,
                              void* d_out, int out_size, void* d_ws,
                              size_t ws_size, hipStream_t stream) {
    (void)in_sizes; (void)n_in; (void)out_size; (void)d_ws; (void)ws_size;
    const float* x = (const float*)d_in[0];
    float* out = (float*)d_out;
    dim3 grid((CROP + TILE - 1) / TILE,   // 12 (oj tiles)
              (CROP + TILE - 1) / TILE,   // 12 (oi tiles)
              8);                         // batch
    backproj_tdm_kernel<<<grid, dim3(256, 1, 1), 0, stream>>>(x, out);
}